// SSResidualUnit_10866267259481
// MI455X (gfx1250) — compile-verified
//
#include <hip/hip_runtime.h>
#include <math.h>

// ---------------------------------------------------------------------------
// CDNA5 WMMA f32 16x16x4 types
// ---------------------------------------------------------------------------
typedef float v2f __attribute__((ext_vector_type(2)));
typedef float v8f __attribute__((ext_vector_type(8)));
typedef int   i32x4 __attribute__((vector_size(16)));   // matches builtin's param

struct F12 { float f[12]; };

#define MT2 128          // gemm M tile (32 rows per wave, 4 waves)
#define MT  64           // conv7 M tile
#define NT  64           // N tile
#define KC  16           // reduction chunk

// ---------------------------------------------------------------------------
// gfx1250 async global->LDS staging (ASYNCcnt path), guarded so compile stays
// green on toolchains without the builtins (falls back to sync staging).
// ---------------------------------------------------------------------------
#if defined(__has_builtin)
#  if __has_builtin(__builtin_amdgcn_global_load_async_to_lds_b32) && \
      __has_builtin(__builtin_amdgcn_global_load_async_to_lds_b128)
#    define HAVE_ASYNC 1
#  endif
#endif
#ifndef HAVE_ASYNC
#  define HAVE_ASYNC 0
#endif

#if HAVE_ASYNC
#  define GPTR32(p)  ((__attribute__((address_space(1))) int*)(p))
#  define GPTR128(p) ((__attribute__((address_space(1))) i32x4*)(p))
#  define LPTR32(p)  ((__attribute__((address_space(3))) int*)(p))
#  define LPTR128(p) ((__attribute__((address_space(3))) i32x4*)(p))
#  if __has_builtin(__builtin_amdgcn_s_wait_asynccnt)
#    define WAIT_ASYNC(n) __builtin_amdgcn_s_wait_asynccnt(n)
#  else
#    define WAIT_ASYNC(n) asm volatile("s_wait_asynccnt %0" :: "i"(n) : "memory")
#  endif
#else
#  define WAIT_ASYNC(n) do {} while (0)
#endif

// ---------------------------------------------------------------------------
// Weight-norm: Wn[co][r] = g[co] * v[co][r] / ||v[co]||
// ---------------------------------------------------------------------------
__global__ __launch_bounds__(256) void wnorm_kernel(
    const float* __restrict__ V, const float* __restrict__ G,
    float* __restrict__ Wn, int R)
{
    int co = blockIdx.x;
    const float* v = V + (size_t)co * R;
    __shared__ float red[256];
    float s = 0.f;
    for (int i = threadIdx.x; i < R; i += 256) { float t = v[i]; s += t * t; }
    red[threadIdx.x] = s;
    __syncthreads();
    for (int off = 128; off > 0; off >>= 1) {
        if (threadIdx.x < off) red[threadIdx.x] += red[threadIdx.x + off];
        __syncthreads();
    }
    float scale = G[co] * rsqrtf(red[0]);
    for (int i = threadIdx.x; i < R; i += 256)
        Wn[(size_t)co * R + i] = v[i] * scale;
}

// ---------------------------------------------------------------------------
// Fused act1d: upsample2 (FIR-12, edge pad) -> snake_beta -> downsample2.
// One block per (channel, batch) row. in_mode 1: X is (B,C,T); 0: X is [C][B*T].
// ---------------------------------------------------------------------------
__global__ __launch_bounds__(256) void act_kernel(
    const float* __restrict__ X, float* __restrict__ Y,
    const float* __restrict__ la, const float* __restrict__ lb,
    F12 filt, int T, int C, int B, int in_mode)
{
    int c = blockIdx.x;
    int b = blockIdx.y;
    __shared__ float xs[4096 + 16];
    __shared__ float ss[8192];

    const float* xrow = in_mode
        ? (X + ((size_t)b * C + c) * T)
        : (X + (size_t)c * ((size_t)B * T) + (size_t)b * T);

    for (int i = threadIdx.x; i < T + 10; i += 256) {
        int t = i - 5;
        t = min(max(t, 0), T - 1);
        xs[i] = xrow[t];
    }
    __syncthreads();

    float a  = expf(la[c]);
    float bb = expf(lb[c]);
    float inv_b = 1.0f / (bb + 1e-9f);

    for (int u = threadIdx.x; u < 2 * T; u += 256) {
        int m = u >> 1;
        float up = 0.f;
        if (u & 1) {
#pragma unroll
            for (int aa = 0; aa < 6; ++aa) up += filt.f[2 * aa] * xs[m + 8 - aa];
        } else {
#pragma unroll
            for (int aa = 0; aa < 6; ++aa) up += filt.f[2 * aa + 1] * xs[m + 7 - aa];
        }
        up *= 2.0f;
        float sn = sinf(up * a);
        ss[u] = up + inv_b * sn * sn;
    }
    __syncthreads();

    float* yrow = Y + (size_t)c * ((size_t)B * T) + (size_t)b * T;
    for (int t = threadIdx.x; t < T; t += 256) {
        float acc = 0.f;
#pragma unroll
        for (int j = 0; j < 12; ++j) {
            int u = 2 * t + j - 5;
            u = min(max(u, 0), 2 * T - 1);
            acc += filt.f[j] * ss[u];
        }
        yrow[t] = acc;
    }
}

// ---------------------------------------------------------------------------
// 1x1-conv GEMM, WMMA f32 16x16x4, 128x64 tile, 4 waves x 32 rows,
// double-buffered async LDS staging.  B panel stored COLUMN-major so each
// fragment is one aligned 64-bit LDS load.
//   mode 0: D[M][N] = A[M][K] * Bm[K][N]
//   mode 1: out(B,C,T) = acc + X(B,C,T)
// ---------------------------------------------------------------------------
__global__ __launch_bounds__(128) void gemm_wmma_f32(
    const float* __restrict__ A, const float* __restrict__ Bm,
    float* __restrict__ D, const float* __restrict__ X,
    int M, int K, int N, int T, int C, int mode)
{
    __shared__ __align__(16) float As[2][MT2][KC + 4];  // row stride 80B (16B mult)
    __shared__ __align__(16) float Bs[2][NT][KC + 2];   // col-major, stride 72B

    int m0 = blockIdx.y * MT2;
    int n0 = blockIdx.x * NT;
    int tid  = threadIdx.x;
    int wave = tid >> 5, lane = tid & 31, half = lane >> 4, ln = lane & 15;

    v8f acc0[4] = {}, acc1[4] = {};

    auto stage = [&](int bf, int k0) {
#if HAVE_ASYNC
#pragma unroll
        for (int q = 0; q < 4; ++q) {                 // A: 512 x 16B chunks
            int e = q * 128 + tid;
            int m = e >> 2, c4 = (e & 3) << 2;
            __builtin_amdgcn_global_load_async_to_lds_b128(
                GPTR128(A + (size_t)(m0 + m) * K + k0 + c4),
                LPTR128(&As[bf][m][c4]), 0, 0);
        }
#pragma unroll
        for (int q = 0; q < 8; ++q) {                 // B: 1024 x b32 (transpose)
            int e = q * 128 + tid;
            int n = e & 63, r = e >> 6;
            __builtin_amdgcn_global_load_async_to_lds_b32(
                GPTR32(Bm + (size_t)(k0 + r) * N + n0 + n),
                LPTR32(&Bs[bf][n][r]), 0, 0);
        }
#else
        for (int e = tid; e < MT2 * KC; e += 128) {
            int m = e >> 4, kk = e & 15;
            As[bf][m][kk] = A[(size_t)(m0 + m) * K + k0 + kk];
        }
        for (int e = tid; e < KC * NT; e += 128) {
            int n = e & 63, r = e >> 6;
            Bs[bf][n][r] = Bm[(size_t)(k0 + r) * N + n0 + n];
        }
#endif
    };

    stage(0, 0);
    int buf = 0;
    for (int k0 = 0; k0 < K; k0 += KC) {
        bool nxt = (k0 + KC) < K;
        if (nxt) { stage(buf ^ 1, k0 + KC); WAIT_ASYNC(12); }
        else     { WAIT_ASYNC(0); }
        __syncthreads();
#pragma unroll
        for (int kk = 0; kk < KC; kk += 4) {
            int r = kk + 2 * half;
            v2f a0 = *(const v2f*)&As[buf][wave * 32 + ln][r];
            v2f a1 = *(const v2f*)&As[buf][wave * 32 + 16 + ln][r];
#pragma unroll
            for (int j = 0; j < 4; ++j) {
                v2f bv = *(const v2f*)&Bs[buf][j * 16 + ln][r];
                acc0[j] = __builtin_amdgcn_wmma_f32_16x16x4_f32(
                    false, a0, false, bv, (short)0, acc0[j], false, false);
                acc1[j] = __builtin_amdgcn_wmma_f32_16x16x4_f32(
                    false, a1, false, bv, (short)0, acc1[j], false, false);
            }
        }
        __syncthreads();
        buf ^= 1;
    }

#pragma unroll
    for (int j = 0; j < 4; ++j) {
#pragma unroll
        for (int v = 0; v < 8; ++v) {
            int n = n0 + j * 16 + ln;
#pragma unroll
            for (int h = 0; h < 2; ++h) {
                int m = m0 + wave * 32 + h * 16 + v + half * 8;
                float val = h ? acc1[j][v] : acc0[j][v];
                if (mode == 1) {
                    int b2 = n / T, t = n - b2 * T;
                    size_t idx = (size_t)b2 * C * T + (size_t)m * T + t;
                    D[idx] = val + X[idx];
                } else {
                    D[(size_t)m * N + n] = val;
                }
            }
        }
    }
}

// ---------------------------------------------------------------------------
// K=7 conv as implicit GEMM (zero pad 3): 64x64 tile, ci chunks of 16,
// col-major halo'd B panel, double-buffered async staging.  OOB halo cells
// are pre-zeroed once; their async copies are redirected to a dummy slot.
// ---------------------------------------------------------------------------
__global__ __launch_bounds__(128) void conv7_wmma_f32(
    const float* __restrict__ W,   // [512][512][7] normalized
    const float* __restrict__ H,   // [C][N]
    float* __restrict__ D,         // [C][N]
    int Cc, int N, int T)
{
    __shared__ __align__(16) float As[2][MT][116];      // 112 used, stride 464B
    __shared__ __align__(16) float Bs[2][72][KC + 2];   // col-major [pos][ci]
    __shared__ float dummy[128];

    int m0 = blockIdx.y * MT;
    int n0 = blockIdx.x * NT;
    int tid  = threadIdx.x;
    int wave = tid >> 5, lane = tid & 31, half = lane >> 4, ln = lane & 15;
    int bb = n0 / T, t0 = n0 - bb * T;

    // pre-zero OOB halo cells (async never writes them)
    if (t0 == 0)
        for (int e = tid; e < 2 * 3 * 18; e += 128) {
            int b2 = e / 54, rem = e % 54;
            Bs[b2][rem / 18][rem % 18] = 0.f;
        }
    if (t0 + NT == T)
        for (int e = tid; e < 2 * 3 * 18; e += 128) {
            int b2 = e / 54, rem = e % 54;
            Bs[b2][67 + rem / 18][rem % 18] = 0.f;
        }

    v8f acc[4] = {};

    auto stage = [&](int bf, int ci0) {
#if HAVE_ASYNC
#pragma unroll
        for (int q = 0; q < 14; ++q) {                // A: 1792 x 16B chunks
            int e = q * 128 + tid;
            int m = e / 28, c4 = (e % 28) << 2;
            __builtin_amdgcn_global_load_async_to_lds_b128(
                GPTR128(W + ((size_t)(m0 + m) * Cc + ci0) * 7 + c4),
                LPTR128(&As[bf][m][c4]), 0, 0);
        }
#pragma unroll
        for (int q = 0; q < 9; ++q) {                 // B: 1152 x b32 (transpose)
            int e = q * 128 + tid;
            int c = e / 72, p = e % 72;
            int t = t0 - 3 + p;
            bool ok = (p < 70) && (t >= 0) && (t < T);
            const float* gsrc = H + (size_t)(ci0 + c) * N + (size_t)bb * T + (ok ? t : 0);
            float* ldst = ok ? &Bs[bf][p][c] : &dummy[tid];
            __builtin_amdgcn_global_load_async_to_lds_b32(GPTR32(gsrc), LPTR32(ldst), 0, 0);
        }
#else
        for (int e = tid; e < MT * 112; e += 128) {
            int m = e / 112, r = e % 112;
            As[bf][m][r] = W[((size_t)(m0 + m) * Cc + ci0) * 7 + r];
        }
        for (int e = tid; e < 16 * 70; e += 128) {
            int c = e / 70, p = e % 70;
            int t = t0 - 3 + p;
            Bs[bf][p][c] = (t >= 0 && t < T)
                ? H[(size_t)(ci0 + c) * N + (size_t)bb * T + t] : 0.f;
        }
#endif
    };

    stage(0, 0);
    int buf = 0;
    for (int ci0 = 0; ci0 < Cc; ci0 += KC) {
        bool nxt = (ci0 + KC) < Cc;
        if (nxt) { stage(buf ^ 1, ci0 + KC); WAIT_ASYNC(23); }
        else     { WAIT_ASYNC(0); }
        __syncthreads();
#pragma unroll
        for (int kk = 0; kk < KC; kk += 4) {
            int r = kk + 2 * half;
#pragma unroll
            for (int k = 0; k < 7; ++k) {
                v2f a;
                a.x = As[buf][wave * 16 + ln][r * 7 + k];
                a.y = As[buf][wave * 16 + ln][(r + 1) * 7 + k];
#pragma unroll
                for (int j = 0; j < 4; ++j) {
                    v2f bv = *(const v2f*)&Bs[buf][j * 16 + ln + k][r];
                    acc[j] = __builtin_amdgcn_wmma_f32_16x16x4_f32(
                        false, a, false, bv, (short)0, acc[j], false, false);
                }
            }
        }
        __syncthreads();
        buf ^= 1;
    }

#pragma unroll
    for (int j = 0; j < 4; ++j) {
#pragma unroll
        for (int v = 0; v < 8; ++v) {
            int m = m0 + wave * 16 + v + half * 8;
            int n = n0 + j * 16 + ln;
            D[(size_t)m * N + n] = acc[j][v];
        }
    }
}

// ---------------------------------------------------------------------------
// Host side
// ---------------------------------------------------------------------------
static double bessel_i0(double x) {
    double s = 1.0, t = 1.0;
    for (int k = 1; k < 64; ++k) { double q = x / (2.0 * k); t *= q * q; s += t; }
    return s;
}

extern "C" void kernel_launch(void* const* d_in, const int* in_sizes, int n_in,
                              void* d_out, int out_size, void* d_ws, size_t ws_size,
                              hipStream_t stream) {
    (void)in_sizes; (void)n_in; (void)out_size; (void)ws_size;
    const int B = 4, C = 512, T = 4096;
    const int N = B * T;
    const int C4 = 4 * C;

    const float* x  = (const float*)d_in[0];
    const float* a1 = (const float*)d_in[1];
    const float* b1 = (const float*)d_in[2];
    const float* v1 = (const float*)d_in[3];
    const float* g1 = (const float*)d_in[4];
    const float* a2 = (const float*)d_in[5];
    const float* b2 = (const float*)d_in[6];
    const float* v2 = (const float*)d_in[7];
    const float* g2 = (const float*)d_in[8];
    const float* v3 = (const float*)d_in[9];
    const float* g3 = (const float*)d_in[10];
    float* out = (float*)d_out;

    // Kaiser-sinc FIR (K=12, cutoff .25, half_width .3) computed on host.
    F12 F;
    {
        const int K = 12;
        const double PI = 3.14159265358979323846;
        const double cutoff = 0.25, hw = 0.3;
        double A = 2.285 * (K / 2 - 1) * PI * (4.0 * hw) + 7.95;
        double beta;
        if (A > 50.0)      beta = 0.1102 * (A - 8.7);
        else if (A >= 21.) beta = 0.5842 * pow(A - 21.0, 0.4) + 0.07886 * (A - 21.0);
        else               beta = 0.0;
        double f[12], sum = 0.0, denom = bessel_i0(beta);
        for (int n = 0; n < K; ++n) {
            double r = (2.0 * n) / (K - 1) - 1.0;
            double win = bessel_i0(beta * sqrt(fmax(0.0, 1.0 - r * r))) / denom;
            double t = (n - K / 2) + 0.5;
            double xx = 2.0 * cutoff * t;
            double sinc = (xx == 0.0) ? 1.0 : sin(PI * xx) / (PI * xx);
            f[n] = 2.0 * cutoff * win * sinc;
            sum += f[n];
        }
        for (int n = 0; n < K; ++n) F.f[n] = (float)(f[n] / sum);
    }

    // Workspace layout (floats)
    float* ws  = (float*)d_ws;
    float* w1n = ws;                          // 512*3584
    float* w2n = w1n + (size_t)C * C * 7;     // 2048*512
    float* w3n = w2n + (size_t)C4 * C;        // 512*2048
    float* h1  = w3n + (size_t)C * C4;        // 512*16384
    float* h2  = h1  + (size_t)C * N;         // 512*16384
    float* h4  = h2  + (size_t)C * N;         // 2048*16384

    wnorm_kernel<<<C,  256, 0, stream>>>(v1, g1, w1n, C * 7);
    wnorm_kernel<<<C4, 256, 0, stream>>>(v2, g2, w2n, C);
    wnorm_kernel<<<C,  256, 0, stream>>>(v3, g3, w3n, C4);

    act_kernel<<<dim3(C, B), 256, 0, stream>>>(x, h1, a1, b1, F, T, C, B, 1);

    conv7_wmma_f32<<<dim3(N / NT, C / MT), 128, 0, stream>>>(w1n, h1, h2, C, N, T);

    act_kernel<<<dim3(C, B), 256, 0, stream>>>(h2, h1, a2, b2, F, T, C, B, 0);

    gemm_wmma_f32<<<dim3(N / NT, C4 / MT2), 128, 0, stream>>>(
        w2n, h1, h4, nullptr, C4, C, N, T, C, 0);

    gemm_wmma_f32<<<dim3(N / NT, C / MT2), 128, 0, stream>>>(
        w3n, h4, out, x, C, C4, N, T, C, 1);
}